// PinUtilization_40948218200501
// MI455X (gfx1250) — compile-verified
//
#include <hip/hip_runtime.h>
#include <stdint.h>

// ---------------------------------------------------------------------------
// Pin utilization map (DREAMPlace-style) for MI455X / gfx1250.
//
// Memory-bound scatter: 40MB streamed reads, ~12-32M f32 atomics into a 1MB
// grid that lives entirely in the 192MB L2. No matmul structure -> no WMMA.
// CDNA5-specific paths used:
//   * global_load_async_to_lds_b32 + s_wait_asynccnt (ASYNCcnt staging)
//   * native non-returning global_atomic_add_f32 (inline asm, avoids any
//     CAS-loop expansion of fp atomics)
//   * wave32 block sizing (256 threads = 8 waves per WGP)
// ---------------------------------------------------------------------------

#define NP   2000000            // NUM_PHYSICAL
#define NN   2500000            // NUM_NODES
#define NBX_ 512
#define NBY_ 512
#define BSXf 1.953125f          // 1000/512, exact in binary (125/64)
#define BSYf 1.953125f
#define HSTRf 1.3810678f        // 0.5 * BSX * sqrt(2)
#define NORMf 61.03515625f      // BSX*BSY*UNIT_PIN_CAPACITY (exact)
#define MINRf 0.4f
#define MAXRf 2.5f

__device__ __forceinline__ unsigned lds_off_u32(const void* p) {
  // Low 32 bits of a generic pointer into LDS == LDS byte offset (flat
  // aperture mapping uses addr[31:0]); this is what the async-LDS VDST wants.
  return (unsigned)(uintptr_t)p;
}

extern "C" __global__ void pinutil_zero_kernel(float* __restrict__ acc, int n) {
  int i = blockIdx.x * blockDim.x + threadIdx.x;
  if (i < n) acc[i] = 0.0f;
}

extern "C" __global__ void __launch_bounds__(256)
PinUtilization_40948218200501_kernel(const float* __restrict__ pos,
                                     const float* __restrict__ nsx,
                                     const float* __restrict__ nsy,
                                     const float* __restrict__ pw,
                                     float* __restrict__ acc)
{
  __shared__ float sX[256], sY[256], sSX[256], sSY[256], sW[256];

  const int tid = (int)threadIdx.x;
  const int gi  = (int)blockIdx.x * 256 + tid;
  const int gl  = (gi < NP) ? gi : (NP - 1);       // clamp tail lanes (loads only)

  const unsigned offA = (unsigned)gl * 4u;          // byte offset: x, sx, sy, w
  const unsigned offY = (unsigned)(gl + NN) * 4u;   // y lives at pos + NUM_NODES

  const unsigned aX  = lds_off_u32(&sX[tid]);
  const unsigned aY  = lds_off_u32(&sY[tid]);
  const unsigned aSX = lds_off_u32(&sSX[tid]);
  const unsigned aSY = lds_off_u32(&sSY[tid]);
  const unsigned aW  = lds_off_u32(&sW[tid]);

  // Async DMA of the 5 input streams into LDS (per-lane gather, ASYNCcnt).
  asm volatile("global_load_async_to_lds_b32 %0, %1, %2" :: "v"(aX),  "v"(offA), "s"(pos) : "memory");
  asm volatile("global_load_async_to_lds_b32 %0, %1, %2" :: "v"(aY),  "v"(offY), "s"(pos) : "memory");
  asm volatile("global_load_async_to_lds_b32 %0, %1, %2" :: "v"(aSX), "v"(offA), "s"(nsx) : "memory");
  asm volatile("global_load_async_to_lds_b32 %0, %1, %2" :: "v"(aSY), "v"(offA), "s"(nsy) : "memory");
  asm volatile("global_load_async_to_lds_b32 %0, %1, %2" :: "v"(aW),  "v"(offA), "s"(pw)  : "memory");
  asm volatile("s_wait_asynccnt 0" ::: "memory");
  // Each lane reads back only its own slots -> no workgroup barrier needed.

  const float x  = sX[tid];
  const float y  = sY[tid];
  const float sx = sSX[tid];
  const float sy = sSY[tid];
  const float w  = sW[tid];

  // hx = 0.5*max(sx, BSX*sqrt2) == max(0.5*sx, 0.5*BSX*sqrt2)
  const float hx = fmaxf(0.5f * sx, HSTRf);
  const float hy = fmaxf(0.5f * sy, HSTRf);
  const float cx = x + 0.5f * sx;
  const float cy = y + 0.5f * sy;
  const float xmin = cx - hx, xmax = cx + hx;
  const float ymin = cy - hy, ymax = cy + hy;
  const float dens = w / (4.0f * hx * hy);

  const int bxl = (int)floorf(xmin / BSXf);
  const int byl = (int)floorf(ymin / BSYf);

  if (gi < NP) {
#pragma unroll
    for (int kx = 0; kx < 4; ++kx) {
      const int ix = bxl + kx;
      const float bxlo = (float)ix * BSXf;
      const float ox = fminf(xmax, bxlo + BSXf) - fmaxf(xmin, bxlo);
      if (ix >= 0 && ix < NBX_ && ox > 0.0f) {
        const float oxd = ox * dens;
#pragma unroll
        for (int ky = 0; ky < 4; ++ky) {
          const int iy = byl + ky;
          const float bylo = (float)iy * BSYf;
          const float oy = fminf(ymax, bylo + BSYf) - fmaxf(ymin, bylo);
          if (iy >= 0 && iy < NBY_ && oy > 0.0f) {
            const unsigned boff = (unsigned)(ix * NBY_ + iy) * 4u;
            const float val = oxd * oy;
            // Non-returning HW f32 atomic add (saddr form), default TH=RT.
            asm volatile("global_atomic_add_f32 %0, %1, %2"
                         :: "v"(boff), "v"(val), "s"(acc) : "memory");
          }
        }
      }
    }
  }
  // s_endpgm performs an implicit wait-idle; atomics drain before wave exit.
}

extern "C" __global__ void pinutil_finalize_kernel(float* __restrict__ acc, int n) {
  int i = blockIdx.x * blockDim.x + threadIdx.x;
  if (i < n) {
    float v = acc[i] / NORMf;
    acc[i] = fminf(fmaxf(v, MINRf), MAXRf);
  }
}

extern "C" void kernel_launch(void* const* d_in, const int* in_sizes, int n_in,
                              void* d_out, int out_size, void* d_ws, size_t ws_size,
                              hipStream_t stream) {
  (void)in_sizes; (void)n_in; (void)d_ws; (void)ws_size;

  const float* pos = (const float*)d_in[0];  // 2*NN floats (x then y)
  const float* nsx = (const float*)d_in[1];  // NN floats
  const float* nsy = (const float*)d_in[2];  // NN floats
  const float* pw  = (const float*)d_in[3];  // NP floats
  float* out = (float*)d_out;                // NBX*NBY floats

  const int nbins = NBX_ * NBY_;             // 262144
  const int zb = (nbins + 255) / 256;        // 1024 blocks
  const int sb = (NP + 255) / 256;           // 7813 blocks

  // Accumulate directly into d_out (it is re-zeroed every call, so the
  // launch is deterministic and graph-replay safe).
  pinutil_zero_kernel<<<zb, 256, 0, stream>>>(out, nbins);
  PinUtilization_40948218200501_kernel<<<sb, 256, 0, stream>>>(pos, nsx, nsy, pw, out);
  pinutil_finalize_kernel<<<zb, 256, 0, stream>>>(out, nbins);
  (void)out_size;
}